// HypergraphModel_66477503807916
// MI455X (gfx1250) — compile-verified
//
#include <hip/hip_runtime.h>
#include <hip/hip_bf16.h>

// Problem constants (match reference)
#define N_NODES 200000
#define N_EDGES 50000
#define NNZ_TOT 1600000
#define IN_CH   128
#define HIDDEN  256

typedef __attribute__((ext_vector_type(2))) float v2f;
typedef __attribute__((ext_vector_type(8))) float v8f;

// ---------------------------------------------------------------------------
// SpMM scatter: x_hyperedges[row] += val * x[col]   (one wave per nnz entry)
// 32 lanes/nnz, each lane handles 4 contiguous channels via float4.
// ---------------------------------------------------------------------------
__global__ void hg_spmm_scatter(const float* __restrict__ x,
                                const int*   __restrict__ inc_rows,
                                const int*   __restrict__ inc_cols,
                                const float* __restrict__ inc_vals,
                                float*       __restrict__ xh,
                                int nnz)
{
    int t = blockIdx.x * blockDim.x + threadIdx.x;
    int e = t >> 5;                 // wave-uniform nnz index
    if (e >= nnz) return;
    int c4 = (t & 31) * 4;          // channel base: lane*4, covers 128 channels

    int   r = inc_rows[e];
    int   c = inc_cols[e];
    float v = inc_vals[e];

    const float4 xv = *(const float4*)(x + (size_t)c * IN_CH + c4);
    float* dst = xh + (size_t)r * IN_CH + c4;
    unsafeAtomicAdd(dst + 0, v * xv.x);
    unsafeAtomicAdd(dst + 1, v * xv.y);
    unsafeAtomicAdd(dst + 2, v * xv.z);
    unsafeAtomicAdd(dst + 3, v * xv.w);
}

// ---------------------------------------------------------------------------
// Fused GEMM + bias + ReLU:  out = relu(A[M,128] @ W[128,256] + b[256])
// One wave per 16x16 output tile, K swept in steps of 4 using
// V_WMMA_F32_16X16X4_F32 (fp32-exact, matches reference precision).
//
// Fragment layouts per CDNA5 ISA (CDNA5 ISA):
//   A 16x4 f32 : lane l<16 -> row M=l, {K=0,K=1}; lane l>=16 -> row M=l-16, {K=2,K=3}
//   B 4x16 f32 : lane l<16 -> col N=l, {K=0,K=1}; lane l>=16 -> col N=l-16, {K=2,K=3}
//   C/D 16x16  : lane l<16 -> VGPR v = (M=v, N=l); lane l>=16 -> (M=v+8, N=l-16)
// ---------------------------------------------------------------------------
__global__ void hg_gemm_bias_relu(const float* __restrict__ A,
                                  const float* __restrict__ W,
                                  const float* __restrict__ bias,
                                  float*       __restrict__ out,
                                  int M)
{
    const int K = IN_CH;     // 128
    const int N = HIDDEN;    // 256

    const int wave = threadIdx.x >> 5;            // 0..7
    const int lane = threadIdx.x & 31;
    const int mt   = blockIdx.x * 8 + wave;       // 16-row tile index
    if (mt * 16 >= M) return;                     // wave-uniform exit

    const int n0   = blockIdx.y * 16;
    const int half = lane >> 4;                   // 0: lanes 0-15, 1: lanes 16-31
    const int l16  = lane & 15;
    const int kb   = half * 2;                    // K sub-offset {0,2}
    const int row  = mt * 16 + l16;               // A row for this lane
    const int col  = n0 + l16;                    // B/D column for this lane

    const float* __restrict__ arow = A + (size_t)row * K;

    v8f acc = {};
    #pragma unroll
    for (int k = 0; k < K; k += 4) {
        // A fragment: two consecutive f32 (even-aligned -> single b64 load)
        v2f a = *(const v2f*)(arow + k + kb);
        // B fragment: two strided loads of the hot 128KB weight (L2-resident)
        v2f b;
        b.x = W[(size_t)(k + kb)     * N + col];
        b.y = W[(size_t)(k + kb + 1) * N + col];
        // 8 args: (neg_a, A, neg_b, B, c_mod, C, reuse_a, reuse_b)
        acc = __builtin_amdgcn_wmma_f32_16x16x4_f32(
                  false, a, false, b, (short)0, acc, false, false);
    }

    // Epilogue: bias + ReLU, store per C/D layout.
    const float bv   = bias[col];
    const int   mrow = mt * 16 + (half ? 8 : 0);
    float* __restrict__ orow = out + (size_t)mrow * N + col;
    #pragma unroll
    for (int v = 0; v < 8; ++v) {
        float r = acc[v] + bv;
        r = r > 0.0f ? r : 0.0f;
        orow[(size_t)v * N] = r;
    }
}

// ---------------------------------------------------------------------------
// Launch
// ---------------------------------------------------------------------------
extern "C" void kernel_launch(void* const* d_in, const int* in_sizes, int n_in,
                              void* d_out, int out_size, void* d_ws, size_t ws_size,
                              hipStream_t stream)
{
    const float* x        = (const float*)d_in[0];
    const int*   inc_rows = (const int*)  d_in[1];
    const int*   inc_cols = (const int*)  d_in[2];
    const float* inc_vals = (const float*)d_in[3];
    const float* W_nodes  = (const float*)d_in[4];
    const float* b_nodes  = (const float*)d_in[5];
    const float* W_edges  = (const float*)d_in[6];
    const float* b_edges  = (const float*)d_in[7];

    float* out_nodes = (float*)d_out;                                  // [N_NODES, HIDDEN]
    float* out_edges = out_nodes + (size_t)N_NODES * HIDDEN;           // [N_EDGES, HIDDEN]
    float* xh        = (float*)d_ws;                                   // [N_EDGES, IN_CH]

    // 1) zero the hyperedge accumulator (graph-capturable memset node)
    hipMemsetAsync(xh, 0, (size_t)N_EDGES * IN_CH * sizeof(float), stream);

    // 2) scatter-add SpMM: 32 lanes per nnz, 256 threads/block -> 8 nnz/block
    {
        int threads = 256;
        int blocks  = (NNZ_TOT * 32 + threads - 1) / threads;          // 200000
        hg_spmm_scatter<<<blocks, threads, 0, stream>>>(
            x, inc_rows, inc_cols, inc_vals, xh, NNZ_TOT);
    }

    // 3) nodes branch: relu(x @ W_nodes + b_nodes)
    {
        int mtiles = N_NODES / 16;                                     // 12500
        dim3 grid((mtiles + 7) / 8, HIDDEN / 16);                      // (1563, 16)
        hg_gemm_bias_relu<<<grid, 256, 0, stream>>>(
            x, W_nodes, b_nodes, out_nodes, N_NODES);
    }

    // 4) edges branch: relu(xh @ W_edges + b_edges)  (after scatter on same stream)
    {
        int mtiles = N_EDGES / 16;                                     // 3125
        dim3 grid((mtiles + 7) / 8, HIDDEN / 16);                      // (391, 16)
        hg_gemm_bias_relu<<<grid, 256, 0, stream>>>(
            xh, W_edges, b_edges, out_edges, N_EDGES);
    }
}